// RNN_21955872817552
// MI455X (gfx1250) — compile-verified
//
#include <hip/hip_runtime.h>
#include <hip/hip_bf16.h>
#include <math.h>

// ---------------------------------------------------------------------------
// SimpleRNN forward for MI455X (gfx1250, wave32, WMMA).
//   Kernel 1: xw = X @ W_xh + b           (memory-bound fp32 WMMA GEMM,
//                                          LDS-transposed coalesced stores)
//   Kernel 2: persistent scan over T with grid-wide barrier:
//             h_t = tanh(xw_t + h_{t-1} @ W_hh)
//             W_hh column slices cached in LDS across all 1024 steps.
//   seq_out buffer holds xw, overwritten in-place by h_t slice by slice.
// ---------------------------------------------------------------------------

typedef __attribute__((ext_vector_type(2))) float v2f;
typedef __attribute__((ext_vector_type(8))) float v8f;

#define RNN_B 32
#define RNN_T 1024
#define RNN_D 512
#define LDS_STRIDE 516           // 512 + 4 pad -> conflict-free ds reads
#define SCAN_NWG 16              // 2 row-tiles x 8 col-groups, all resident

// -------------------- Kernel 1: input projection GEMM ----------------------
// Grid: (2048, 4), block 256 (8 waves). WG covers 16 rows x 128 cols.
// A (the 64MB X stream) staged coalesced in LDS; B (W_xh, 1MB) from L2.
// Output transposed through LDS for full-cacheline global_store_b128.
__global__ __launch_bounds__(256) void rnn_proj_kernel(
    const float* __restrict__ X, const float* __restrict__ Wxh,
    const float* __restrict__ bias, float* __restrict__ out,
    unsigned* __restrict__ counter)
{
    if (blockIdx.x == 0 && blockIdx.y == 0 && threadIdx.x == 0)
        *counter = 0u;   // reset grid-barrier counter for the scan kernel

    __shared__ __align__(16) float A[16][LDS_STRIDE];
    __shared__ __align__(16) float O[16][132];      // 16 x 128 + pad

    const int tid  = threadIdx.x;
    const int wave = tid >> 5;
    const int lane = tid & 31;
    const int half = lane >> 4;       // half-wave: K phase (A/B), M phase (C)
    const int ln   = lane & 15;       // row (A) / col (B,C) within tile
    const int m0   = blockIdx.x * 16;              // 0..32767 step 16
    const int ng0  = blockIdx.y * 128;             // column-group base
    const int n0   = ng0 + wave * 16;              // this wave's 16 columns

    // Stage A tile rows [m0, m0+16) x K=512, coalesced float4 loads.
    #pragma unroll
    for (int it = 0; it < 8; ++it) {
        int idx = it * 256 + tid;       // 0..2047 = 16 rows * 128 float4
        int row = idx >> 7;
        int c4  = idx & 127;
        float4 v = reinterpret_cast<const float4*>(X + (size_t)(m0 + row) * RNN_D)[c4];
        reinterpret_cast<float4*>(&A[row][0])[c4] = v;
    }
    __syncthreads();

    v8f c = {};
    #pragma unroll 8
    for (int k = 0; k < RNN_D; k += 4) {
        const int kk = k + 2 * half;
        v2f a = *reinterpret_cast<const v2f*>(&A[ln][kk]);   // A[ln][kk..kk+1]
        v2f b;
        b.x = Wxh[(size_t)kk * RNN_D + n0 + ln];             // B[kk][n]
        b.y = Wxh[(size_t)(kk + 1) * RNN_D + n0 + ln];       // B[kk+1][n]
        c = __builtin_amdgcn_wmma_f32_16x16x4_f32(
                false, a, false, b, (short)0, c, false, false);
    }

    // Transpose through LDS -> coalesced 128B stores of the 64MB stream.
    const float bb = bias[n0 + ln];
    #pragma unroll
    for (int i = 0; i < 8; ++i)
        O[i + 8 * half][wave * 16 + ln] = c[i] + bb;   // M = i+8*half, N = ln
    __syncthreads();

    #pragma unroll
    for (int it = 0; it < 2; ++it) {
        int idx = it * 256 + tid;       // 0..511 = 16 rows * 32 float4
        int row = idx >> 5;
        int c4  = idx & 31;
        float4 v = *reinterpret_cast<const float4*>(&O[row][c4 * 4]);
        reinterpret_cast<float4*>(out + (size_t)(m0 + row) * RNN_D + ng0)[c4] = v;
    }
}

// -------------------- Kernel 2: persistent recurrent scan ------------------
// Grid: 16 WGs (must all be resident), block 128 (4 waves).
// WG (mtile, ngrp): rows = batches [mtile*16, +16), cols [ngrp*64, +64).
// W_hh slice (512x64) cached transposed in LDS ONCE for all 1024 steps;
// per step: stage h_{t-1} (16x512) into LDS, 128 fp32 WMMAs per wave
// fed purely by ds_load_b64, tanh-combine with xw in-place, grid barrier.
__global__ __launch_bounds__(128) void rnn_scan_kernel(
    const float* __restrict__ Whh, float* __restrict__ out,
    float* __restrict__ hfin, unsigned* __restrict__ counter)
{
    __shared__ __align__(16) float A[16][LDS_STRIDE];            // 33 KB
    __shared__ __align__(16) float Bw[4][16][LDS_STRIDE];        // 132 KB

    const int tid  = threadIdx.x;
    const int wave = tid >> 5;
    const int lane = tid & 31;
    const int half = lane >> 4;
    const int ln   = lane & 15;
    const int mtile = blockIdx.x & 1;        // 2 batch-row tiles
    const int ngrp  = blockIdx.x >> 1;       // 8 column groups of 64
    const int m0    = mtile * 16;
    const int nbase = ngrp * 64;             // WG's 64 columns
    const int nc    = nbase + wave * 16 + ln;  // this lane's output column

    // Cache W_hh[:, nbase..nbase+63] transposed: Bw[w][n%16][k] = Whh[k][n].
    for (int it = 0; it < 64; ++it) {
        int idx = it * 128 + tid;            // 0..8191 = 512 k * 16 float4
        int k   = idx >> 4;
        int c4  = idx & 15;                  // float4 within the 64 columns
        float4 v = reinterpret_cast<const float4*>(Whh + (size_t)k * RNN_D + nbase)[c4];
        int n = c4 * 4;
        Bw[(n    ) >> 4][(n    ) & 15][k] = v.x;
        Bw[(n + 1) >> 4][(n + 1) & 15][k] = v.y;
        Bw[(n + 2) >> 4][(n + 2) & 15][k] = v.z;
        Bw[(n + 3) >> 4][(n + 3) & 15][k] = v.w;
    }
    __syncthreads();

    for (int t = 0; t < RNN_T; ++t) {
        v8f c = {};
        if (t > 0) {
            // Stage h_{t-1}: batch rows m0..m0+15, slice t-1 of seq_out.
            #pragma unroll
            for (int j = 0; j < 16; ++j) {
                float4 v = reinterpret_cast<const float4*>(
                    out + ((size_t)(m0 + j) * RNN_T + (t - 1)) * RNN_D)[tid];
                reinterpret_cast<float4*>(&A[j][0])[tid] = v;
            }
            __syncthreads();

            #pragma unroll 8
            for (int k = 0; k < RNN_D; k += 4) {
                const int kk = k + 2 * half;
                v2f a = *reinterpret_cast<const v2f*>(&A[ln][kk]);
                v2f b = *reinterpret_cast<const v2f*>(&Bw[wave][ln][kk]);
                c = __builtin_amdgcn_wmma_f32_16x16x4_f32(
                        false, a, false, b, (short)0, c, false, false);
            }
        }

        // h_t = tanh(xw_t + h_{t-1} @ W_hh), overwriting xw slice in place.
        #pragma unroll
        for (int i = 0; i < 8; ++i) {
            int bidx = m0 + i + 8 * half;
            size_t off = ((size_t)bidx * RNN_T + t) * RNN_D + nc;
            out[off] = tanhf(c[i] + out[off]);
        }

        // ---- grid-wide barrier (all 16 WGs resident) ----
        __threadfence();          // release h_t device-wide
        __syncthreads();          // all waves in WG done before signaling
        if (tid == 0) {
            const unsigned target = (unsigned)(SCAN_NWG * (t + 1));
            atomicAdd(counter, 1u);
            while (__hip_atomic_load(counter, __ATOMIC_ACQUIRE,
                                     __HIP_MEMORY_SCOPE_AGENT) < target) {
                __builtin_amdgcn_s_sleep(1);
            }
        }
        __syncthreads();
        __threadfence();          // acquire other WGs' h_t
    }

    // h_final = seq_out[:, T-1, :]
    for (int i = blockIdx.x * 128 + tid; i < RNN_B * RNN_D; i += SCAN_NWG * 128) {
        int b = i / RNN_D, d = i % RNN_D;
        hfin[i] = out[((size_t)b * RNN_T + (RNN_T - 1)) * RNN_D + d];
    }
}

// ---------------------------------------------------------------------------
extern "C" void kernel_launch(void* const* d_in, const int* in_sizes, int n_in,
                              void* d_out, int out_size, void* d_ws, size_t ws_size,
                              hipStream_t stream)
{
    const float* X    = (const float*)d_in[0];  // [32,1024,512]
    const float* Wxh  = (const float*)d_in[1];  // [512,512]
    const float* Whh  = (const float*)d_in[2];  // [512,512]
    const float* bias = (const float*)d_in[3];  // [512]

    float* seq_out = (float*)d_out;                                   // [32,1024,512]
    float* h_final = seq_out + (size_t)RNN_B * RNN_T * RNN_D;          // [32,512]
    unsigned* counter = (unsigned*)d_ws;                               // 4 bytes

    dim3 pgrid(2048, 4, 1);   // 32768/16 row tiles x 4 col groups of 128
    rnn_proj_kernel<<<pgrid, 256, 0, stream>>>(X, Wxh, bias, seq_out, counter);

    rnn_scan_kernel<<<SCAN_NWG, 128, 0, stream>>>(Whh, seq_out, h_final, counter);
}